// ScModel_16922171146252
// MI455X (gfx1250) — compile-verified
//
#include <hip/hip_runtime.h>
#include <math.h>

typedef __attribute__((ext_vector_type(2))) float v2f;
typedef __attribute__((ext_vector_type(8))) float v8f;

static constexpr int kN      = 128;
static constexpr int kPlane  = kN * kN;       // 16384
static constexpr int kB      = 16;
static constexpr int kNCoeff = 217;
static constexpr int kLdsStr = 132;           // 16x128 strip, padded stride (bank-conflict free)
static constexpr int kBPT    = 4;             // blocks per transform (2 waves x 16 rows = 32 rows/block)

// per-image slab layout (floats)
static constexpr int OFF_XFR  = 0;
static constexpr int OFF_XFI  = kPlane;
static constexpr int OFF_U1   = 2  * kPlane;  // 16 real planes (j1 in {0,1})
static constexpr int OFF_U1FR = 18 * kPlane;  // 16 complex-real planes
static constexpr int OFF_U1FI = 34 * kPlane;  // 16 complex-imag planes
static constexpr int kSlabFloats = 50 * kPlane;

__device__ __forceinline__ v8f wmma4(v2f a, v2f b, v8f c) {
  // V_WMMA_F32_16X16X4_F32 : D = A(16x4) * B(4x16) + C(16x16), all f32, wave32
  return __builtin_amdgcn_wmma_f32_16x16x4_f32(false, a, false, b, (short)0, c,
                                               false, false);
}

// A fragment: A[m,k] = M[row0+m][kk+k], 16x4 f32 layout:
// lane (0..31): m = lane&15, h = lane>>4; VGPR0 = K=2h, VGPR1 = K=2h+1
__device__ __forceinline__ v2f load_a(const float* M, int ld, int row0, int kk,
                                      int lm, int hh) {
  const float* p = M + (size_t)(row0 + lm) * ld + (kk + 2 * hh);
  v2f r;
  r.x = p[0];
  r.y = p[1];
  return r;
}

// B fragment: B[k,n] = M[kk+k][col0+n]; lane: n = lane&15, h = lane>>4;
// VGPR0 = K=2h row, VGPR1 = K=2h+1 row
__device__ __forceinline__ v2f load_b(const float* M, int ld, int kk, int col0,
                                      int ln, int hh) {
  v2f r;
  r.x = M[(size_t)(kk + 2 * hh) * ld + col0 + ln];
  r.y = M[(size_t)(kk + 2 * hh + 1) * ld + col0 + ln];
  return r;
}

// ---------------------------------------------------------------------------
// init: DFT matrix (Fr=cos, Fi=-sin => forward sigma=+1, inverse sigma=-1),
// 8x8 fold kernel gr from phi_f, zero coefficient accumulator.
// ---------------------------------------------------------------------------
__global__ void init_kernel(const float* __restrict__ phi, float* __restrict__ Fr,
                            float* __restrict__ Fi, float* __restrict__ gr,
                            float* __restrict__ coeffs, int nCoeff) {
  int i = blockIdx.x * blockDim.x + threadIdx.x;
  if (i < kPlane) {
    int n = i >> 7, k = i & 127;
    int p = (n * k) & 127;  // exact phase index
    float th = (float)p * (6.28318530717958647692f / 128.0f);
    Fr[i] = cosf(th);
    Fi[i] = -sinf(th);
  }
  if (i < 64) {
    int a = i >> 3, b = i & 7;
    float s = 0.f;
    for (int p = 0; p < 8; ++p)
      for (int q = 0; q < 8; ++q) {
        int ph = (p * a + q * b) & 7;
        float f = (float)ph * (6.28318530717958647692f / 8.0f);
        size_t idx = ((size_t)(16 * p) * kN + 16 * q) * 2;
        // Re(Phi * e^{-i f}) = PhiR*cos(f) + PhiI*sin(f)
        s += phi[idx] * cosf(f) + phi[idx + 1] * sinf(f);
      }
    gr[i] = s * (1.0f / 16384.0f);
  }
  if (i < nCoeff) coeffs[i] = 0.f;
}

// ---------------------------------------------------------------------------
// order 0: coeff[b,0] = sum_{x,y} x[b,x,y] * gr[x&7, y&7]
// ---------------------------------------------------------------------------
__global__ void order0_kernel(const float* __restrict__ x,
                              const float* __restrict__ gr,
                              float* __restrict__ coeffs) {
  int b = blockIdx.x;
  float s = 0.f;
  for (int i = threadIdx.x; i < kPlane; i += blockDim.x) {
    int r = i >> 7, c = i & 127;
    s += x[(size_t)b * kPlane + i] * gr[((r & 7) << 3) | (c & 7)];
  }
  __shared__ float red[256];
  red[threadIdx.x] = s;
  __syncthreads();
  for (int st = blockDim.x >> 1; st > 0; st >>= 1) {
    if ((int)threadIdx.x < st) red[threadIdx.x] += red[threadIdx.x + st];
    __syncthreads();
  }
  if (threadIdx.x == 0) coeffs[b * kNCoeff + 0] = red[0];
}

// ---------------------------------------------------------------------------
// Workhorse: 2D DFT via WMMA matmuls, Y = F^sigma * Z * F^sigma.
// STAGE 0 (A): x[b] (real)           -> XF[b] complex        (forward)
// STAGE 1 (B): XF[b] * psi[j1,l1]    -> u1 = |.| (+coeff)    (inverse)
// STAGE 2 (C): u1 (real)             -> U1F complex          (forward)
// STAGE 3 (D): U1F[j1,l1]*psi[j2,l2] -> coeff from |.|       (inverse)
// Templated on STAGE so all mode flags are compile-time: branch-free K-loops.
// block = 64 threads = 2 waves; each wave owns one 16-row strip; 4 blocks/transform.
// F-matrix A-fragments for the strip are register-cached across the whole pass.
// ---------------------------------------------------------------------------
template <int STAGE>
__global__ void __launch_bounds__(64, 1)
fft2_stage_kernel(int b0, const float* __restrict__ x,
                  const float* __restrict__ psi, const float* __restrict__ Fr,
                  const float* __restrict__ Fi, const float* __restrict__ gr,
                  float* __restrict__ slabs, float* __restrict__ coeffs) {
  constexpr bool HAS_IMAG = (STAGE == 1) || (STAGE == 3);
  constexpr bool HAS_FILT = (STAGE == 1) || (STAGE == 3);
  constexpr bool HAS_DST  = (STAGE == 0) || (STAGE == 2);
  constexpr bool HAS_COEF = (STAGE == 1) || (STAGE == 3);
  constexpr float SGN   = HAS_FILT ? -1.0f : 1.0f;       // inverse vs forward
  constexpr float SCALE = HAS_FILT ? (1.0f / 16384.0f) : 1.0f;

  __shared__ float sTr[2][16 * kLdsStr];
  __shared__ float sTi[2][16 * kLdsStr];

  const int tId = blockIdx.x / kBPT;
  const int sub = blockIdx.x % kBPT;
  const int wv = threadIdx.x >> 5;
  const int lane = threadIdx.x & 31;
  const int lm = lane & 15;   // M (A-frag) or N (B/C/D frag)
  const int hh = lane >> 4;   // half-wave selector
  const int strip = sub * 2 + wv;
  const int row0 = strip * 16;

  const float* srcR = nullptr;
  const float* srcI = nullptr;
  const float* flt = nullptr;
  float* dstR = nullptr;
  float* dstI = nullptr;
  float* ufield = nullptr;
  int coeffIdx = -1;

  if (STAGE == 0) {  // forward FFT of x[b]
    int li = tId;
    float* slab = slabs + (size_t)li * kSlabFloats;
    srcR = x + (size_t)(b0 + li) * kPlane;
    dstR = slab + OFF_XFR;
    dstI = slab + OFF_XFI;
  } else if (STAGE == 1) {  // u1 = |ifft2(XF * psi[j1,l1])|
    int li = tId / 24, ch = tId % 24;
    int j1 = ch >> 3;
    float* slab = slabs + (size_t)li * kSlabFloats;
    srcR = slab + OFF_XFR;
    srcI = slab + OFF_XFI;
    flt = psi + (size_t)ch * kPlane * 2;
    if (j1 < 2) ufield = slab + OFF_U1 + (size_t)ch * kPlane;
    coeffIdx = (b0 + li) * kNCoeff + 1 + ch;
  } else if (STAGE == 2) {  // forward FFT of u1
    int li = tId / 16, ch = tId % 16;
    float* slab = slabs + (size_t)li * kSlabFloats;
    srcR = slab + OFF_U1 + (size_t)ch * kPlane;
    dstR = slab + OFF_U1FR + (size_t)ch * kPlane;
    dstI = slab + OFF_U1FI + (size_t)ch * kPlane;
  } else {  // order 2
    int li = tId / 192, r = tId % 192;
    int pair = r >> 6, l1 = (r >> 3) & 7, l2 = r & 7;
    int j1 = (pair == 2) ? 1 : 0;
    int j2 = (pair == 0) ? 1 : 2;
    float* slab = slabs + (size_t)li * kSlabFloats;
    srcR = slab + OFF_U1FR + (size_t)(j1 * 8 + l1) * kPlane;
    srcI = slab + OFF_U1FI + (size_t)(j1 * 8 + l1) * kPlane;
    flt = psi + (size_t)(j2 * 8 + l2) * kPlane * 2;
    coeffIdx = (b0 + li) * kNCoeff + 25 + pair * 64 + l1 * 8 + l2;
  }

  // ---- register-cache the strip's F A-fragments (invariant across t) ----
  v2f fA[32], fB[32];
#pragma unroll
  for (int s = 0; s < 32; ++s) {
    fA[s] = load_a(Fr, kN, row0, s * 4, lm, hh);
    fB[s] = load_a(Fi, kN, row0, s * 4, lm, hh);
  }

  // -------- pass 1: T = F^sigma * Z  (rows [row0,row0+16), all cols) --------
#pragma unroll 1
  for (int t = 0; t < 8; ++t) {  // column block
    v8f a1 = (v8f)0.0f, a2 = (v8f)0.0f, a3 = (v8f)0.0f, a4 = (v8f)0.0f;
    const int col0 = t * 16;
#pragma unroll
    for (int s = 0; s < 32; ++s) {
      const int kk = s * 4;
      // B fragment of Z (with optional complex pointwise filter)
      const size_t i0 = (size_t)(kk + 2 * hh) * kN + col0 + lm;
      const size_t i1 = i0 + kN;
      float zr0 = srcR[i0];
      float zr1 = srcR[i1];
      float zi0 = 0.f, zi1 = 0.f;
      if (HAS_IMAG) {
        zi0 = srcI[i0];
        zi1 = srcI[i1];
      }
      if (HAS_FILT) {
        float pr0 = flt[2 * i0], pi0 = flt[2 * i0 + 1];
        float pr1 = flt[2 * i1], pi1 = flt[2 * i1 + 1];
        float t0 = zr0 * pr0 - zi0 * pi0;
        zi0 = zr0 * pi0 + zi0 * pr0;
        zr0 = t0;
        float t1 = zr1 * pr1 - zi1 * pi1;
        zi1 = zr1 * pi1 + zi1 * pr1;
        zr1 = t1;
      }
      v2f br, bi;
      br.x = zr0; br.y = zr1;
      bi.x = zi0; bi.y = zi1;
      a1 = wmma4(fA[s], br, a1);          // Fr*Zr
      a4 = wmma4(fB[s], br, a4);          // Fi*Zr
      if (HAS_IMAG || HAS_FILT) {
        a2 = wmma4(fB[s], bi, a2);        // Fi*Zi
        a3 = wmma4(fA[s], bi, a3);        // Fr*Zi
      }
    }
#pragma unroll
    for (int e = 0; e < 8; ++e) {
      int lr = e + 8 * hh;                // local row in strip
      int gc = col0 + lm;
      sTr[wv][lr * kLdsStr + gc] = a1[e] - SGN * a2[e];
      sTi[wv][lr * kLdsStr + gc] = a3[e] + SGN * a4[e];
    }
  }

  // -------- pass 2: Y = T * F^sigma  (same rows, all col blocks) --------
  float csum = 0.f;
#pragma unroll 1
  for (int t = 0; t < 8; ++t) {
    v8f a1 = (v8f)0.0f, a2 = (v8f)0.0f, a3 = (v8f)0.0f, a4 = (v8f)0.0f;
    const int col0 = t * 16;
#pragma unroll 8
    for (int kk = 0; kk < kN; kk += 4) {
      v2f tr, ti;
      {
        int base = lm * kLdsStr + kk + 2 * hh;
        tr.x = sTr[wv][base];
        tr.y = sTr[wv][base + 1];
        ti.x = sTi[wv][base];
        ti.y = sTi[wv][base + 1];
      }
      v2f br = load_b(Fr, kN, kk, col0, lm, hh);
      v2f bi = load_b(Fi, kN, kk, col0, lm, hh);
      a1 = wmma4(tr, br, a1);             // Tr*Fr
      a2 = wmma4(ti, bi, a2);             // Ti*Fi
      a3 = wmma4(tr, bi, a3);             // Tr*Fi
      a4 = wmma4(ti, br, a4);             // Ti*Fr
    }
#pragma unroll
    for (int e = 0; e < 8; ++e) {
      int row = row0 + e + 8 * hh;
      int col = col0 + lm;
      size_t idx = (size_t)row * kN + col;
      float yr = (a1[e] - SGN * a2[e]) * SCALE;
      float yi = (SGN * a3[e] + a4[e]) * SCALE;
      if (HAS_DST) {
        dstR[idx] = yr;
        dstI[idx] = yi;
      }
      if (HAS_COEF) {
        float u = sqrtf(yr * yr + yi * yi);
        if (ufield) ufield[idx] = u;
        csum += u * gr[((row & 7) << 3) | (col & 7)];
      }
    }
  }

  if (HAS_COEF) {
    for (int off = 16; off > 0; off >>= 1) csum += __shfl_down(csum, off, 32);
    if (lane == 0) atomicAdd(&coeffs[coeffIdx], csum);
  }
}

// ---------------------------------------------------------------------------
// linear layer: out[b,o] = bias[o] + sum_c coeffs[b,c] * weight[o,c]
// ---------------------------------------------------------------------------
__global__ void linear_kernel(const float* __restrict__ coeffs,
                              const float* __restrict__ w,
                              const float* __restrict__ bias,
                              float* __restrict__ out) {
  int t = threadIdx.x;
  if (t >= kB * 6) return;
  int b = t / 6, o = t % 6;
  float s = bias[o];
  for (int c = 0; c < kNCoeff; ++c)
    s += coeffs[b * kNCoeff + c] * w[o * kNCoeff + c];
  out[t] = s;
}

extern "C" void kernel_launch(void* const* d_in, const int* in_sizes, int n_in,
                              void* d_out, int out_size, void* d_ws,
                              size_t ws_size, hipStream_t stream) {
  (void)in_sizes; (void)n_in; (void)out_size;
  const float* x = (const float*)d_in[0];       // (16,128,128) f32
  const float* weight = (const float*)d_in[1];  // (6,217) f32
  const float* bias = (const float*)d_in[2];    // (6,) f32
  const float* phi = (const float*)d_in[3];     // (128,128) c64 interleaved
  const float* psi = (const float*)d_in[4];     // (3,8,128,128) c64 interleaved
  float* out = (float*)d_out;                   // (16,6) f32

  float* ws = (float*)d_ws;
  float* Fr = ws;
  float* Fi = Fr + kPlane;
  float* gr = Fi + kPlane;
  float* coeffs = gr + 64;
  float* slabs = coeffs + kB * kNCoeff;

  size_t fixedFloats = (size_t)2 * kPlane + 64 + kB * kNCoeff;
  size_t wsFloats = ws_size / sizeof(float);
  size_t avail = (wsFloats > fixedFloats) ? (wsFloats - fixedFloats) : 0;
  int chunk = (int)(avail / (size_t)kSlabFloats);
  if (chunk < 1) chunk = 1;
  if (chunk > kB) chunk = kB;

  init_kernel<<<(kPlane + 255) / 256, 256, 0, stream>>>(phi, Fr, Fi, gr, coeffs,
                                                        kB * kNCoeff);
  order0_kernel<<<kB, 256, 0, stream>>>(x, gr, coeffs);

  for (int b0 = 0; b0 < kB; b0 += chunk) {
    int nb = kB - b0;
    if (nb > chunk) nb = chunk;
    fft2_stage_kernel<0><<<nb * 1 * kBPT, 64, 0, stream>>>(b0, x, psi, Fr, Fi,
                                                           gr, slabs, coeffs);
    fft2_stage_kernel<1><<<nb * 24 * kBPT, 64, 0, stream>>>(b0, x, psi, Fr, Fi,
                                                            gr, slabs, coeffs);
    fft2_stage_kernel<2><<<nb * 16 * kBPT, 64, 0, stream>>>(b0, x, psi, Fr, Fi,
                                                            gr, slabs, coeffs);
    fft2_stage_kernel<3><<<nb * 192 * kBPT, 64, 0, stream>>>(b0, x, psi, Fr, Fi,
                                                             gr, slabs, coeffs);
  }

  linear_kernel<<<1, 128, 0, stream>>>(coeffs, weight, bias, out);
}